// SetCriterion_37014028157062
// MI455X (gfx1250) — compile-verified
//
#include <hip/hip_runtime.h>
#include <math.h>
#include <stdint.h>

// Problem constants (match reference)
#define NB 64      // batches
#define NQ 900     // queries
#define NM 64      // max gt per batch
#define NC 4       // classes
#define QP 912     // queries padded to 57*16 for WMMA tiling
#define NCOLP 904  // 901 (cols + sentinel) padded
#define NT 256     // threads per block (8 wave32 waves)

typedef __attribute__((ext_vector_type(2))) float v2f;
typedef __attribute__((ext_vector_type(4))) float v4f;
typedef __attribute__((ext_vector_type(8))) float v8f;
typedef __attribute__((ext_vector_type(4))) int v4i;

// Pointer types matching the async-to-LDS builtin's expected parameters:
// arg0: global (AS1) int4*, arg1: LDS (AS3) int4*
typedef __attribute__((address_space(1))) v4i GV4;
typedef __attribute__((address_space(3))) v4i LV4;

struct SM {
  // fp64 Hungarian solver state (mirrors reference float64 math)
  double u[NM + 1];
  double v[NCOLP];
  double minv[NCOLP];
  double red_val[NT];
  // fp32 data tiles
  float costT[NM][QP];   // [gt row][query col] == reference cost.T
  float prob[QP][NC];    // softmax probs (pad rows zero)
  float pb[QP][6];       // pred boxes (pad rows zero)
  float gb[NM][6];       // compacted valid gt boxes (pad rows zero)
  float bneg[NC][NM];    // -onehot(labels): B matrix of the class-cost GEMM
  int p[NCOLP];          // col -> matched row (1-based), 0 = free
  int way[NCOLP];
  int red_idx[NT];
  int labels_c[NM];      // compacted labels
  unsigned char used[NCOLP];
  int m_valid;
  int j0, i0, done;
};

__device__ __forceinline__ void async_copy_b128(const void* g, void* l) {
#if __has_builtin(__builtin_amdgcn_global_load_async_to_lds_b128)
  // CDNA5 async global -> LDS DMA (tracked by ASYNCcnt)
  __builtin_amdgcn_global_load_async_to_lds_b128(
      (GV4*)(unsigned long long)g,
      (LV4*)(unsigned)(unsigned long long)l, 0, 0);
#else
  *(v4f*)l = *(const v4f*)g;  // plain 128-bit copy fallback
#endif
}

__device__ __forceinline__ void wait_async_all() {
#if __has_builtin(__builtin_amdgcn_s_wait_asynccnt)
  __builtin_amdgcn_s_wait_asynccnt(0);
#else
  asm volatile("s_wait_asynccnt 0" ::: "memory");
#endif
}

__global__ void init_acc_kernel(double* acc) {
  acc[0] = 0.0;  // sum cross-entropy
  acc[1] = 0.0;  // sum |l1|
  acc[2] = 0.0;  // pair count
}

__global__ __launch_bounds__(NT) void match_loss_kernel(
    const float* __restrict__ pred_boxes,    // [B,Q,6]
    const float* __restrict__ pred_classes,  // [B,Q,4]
    const float* __restrict__ gt_boxes,      // [B,M,6]
    const int* __restrict__ gt_labels,       // [B,M]
    const int* __restrict__ gt_valid,        // [B,M]
    double* __restrict__ acc) {
  extern __shared__ char smem_raw[];
  SM* s = (SM*)smem_raw;
  const int b = blockIdx.x;
  const int tid = threadIdx.x;

  // ---------------- Stage pred data into LDS via async DMA ----------------
  const float* pc = pred_classes + (size_t)b * NQ * NC;  // 3600 floats
  const float* pg = pred_boxes + (size_t)b * NQ * 6;     // 5400 floats
  for (int i = tid; i < (NQ * NC) / 4; i += NT)          // 900 x 16B
    async_copy_b128(pc + i * 4, &s->prob[0][0] + i * 4);
  for (int i = tid; i < (NQ * 6) / 4; i += NT)           // 1350 x 16B
    async_copy_b128(pg + i * 4, &s->pb[0][0] + i * 4);

  // Overlap: thread 0 compacts valid GTs while DMA is in flight
  if (tid == 0) {
    const int* lab = gt_labels + b * NM;
    const int* val = gt_valid + b * NM;
    const float* gbg = gt_boxes + (size_t)b * NM * 6;
    int cnt = 0;
    for (int mi = 0; mi < NM; ++mi) {
      if (val[mi]) {
        s->labels_c[cnt] = lab[mi];
        for (int d = 0; d < 6; ++d) s->gb[cnt][d] = gbg[mi * 6 + d];
        ++cnt;
      }
    }
    for (int r = cnt; r < NM; ++r) {
      s->labels_c[r] = 0;
      for (int d = 0; d < 6; ++d) s->gb[r][d] = 0.f;
    }
    s->m_valid = cnt;
  }
  // zero one-hot B matrix and padded tile rows (regions untouched by DMA)
  for (int i = tid; i < NC * NM; i += NT) (&s->bneg[0][0])[i] = 0.f;
  for (int q = NQ + tid; q < QP; q += NT) {
    for (int k = 0; k < NC; ++k) s->prob[q][k] = 0.f;
    for (int d = 0; d < 6; ++d) s->pb[q][d] = 0.f;
  }
  wait_async_all();
  __syncthreads();

  const int n = s->m_valid;

  // in-place softmax over classes; build -onehot(labels) as GEMM B matrix
  for (int q = tid; q < NQ; q += NT) {
    float l0 = s->prob[q][0], l1 = s->prob[q][1], l2 = s->prob[q][2],
          l3 = s->prob[q][3];
    float mx = fmaxf(fmaxf(l0, l1), fmaxf(l2, l3));
    float e0 = expf(l0 - mx), e1 = expf(l1 - mx), e2 = expf(l2 - mx),
          e3 = expf(l3 - mx);
    float inv = 1.f / (e0 + e1 + e2 + e3);
    s->prob[q][0] = e0 * inv;
    s->prob[q][1] = e1 * inv;
    s->prob[q][2] = e2 * inv;
    s->prob[q][3] = e3 * inv;
  }
  for (int j = tid; j < n; j += NT) s->bneg[s->labels_c[j]][j] = -1.0f;
  __syncthreads();

  // -------- Cost matrix: WMMA f32 16x16x4 GEMM (class) fused with box L1 in C
  {
    const int wid = tid >> 5;
    const int lane = tid & 31;
    const int mrow = lane & 15;       // A row / B-&-C column within tile
    const int khalf = lane >> 4;      // K/M half select per ISA f32 layouts
    for (int t = wid; t < 57 * 4; t += NT / 32) {
      const int q0 = (t >> 2) * 16;   // query tile base (M)
      const int g0 = (t & 3) * 16;    // gt tile base (N)
      const int gcol = g0 + mrow;
      v2f a, bb;
      a.x = s->prob[q0 + mrow][khalf * 2 + 0];
      a.y = s->prob[q0 + mrow][khalf * 2 + 1];
      bb.x = s->bneg[khalf * 2 + 0][gcol];
      bb.y = s->bneg[khalf * 2 + 1][gcol];
      v8f c;
#pragma unroll
      for (int r = 0; r < 8; ++r) {   // C layout: M = r + khalf*8, N = mrow
        const int qq = q0 + r + khalf * 8;
        float l1 = 0.f;
#pragma unroll
        for (int d = 0; d < 6; ++d) l1 += fabsf(s->pb[qq][d] - s->gb[gcol][d]);
        c[r] = 5.0f * l1;             // COST_BOX
      }
      v8f dres;
#if __has_builtin(__builtin_amdgcn_wmma_f32_16x16x4_f32)
      dres = __builtin_amdgcn_wmma_f32_16x16x4_f32(false, a, false, bb,
                                                   (short)0, c, false, false);
#else
      // scalar fallback: D = A*B + C with the same lane->element mapping
#pragma unroll
      for (int r = 0; r < 8; ++r) {
        const int qq = q0 + r + khalf * 8;
        float accv = c[r];
        for (int k = 0; k < NC; ++k) accv += s->prob[qq][k] * s->bneg[k][gcol];
        dres[r] = accv;
      }
#endif
#pragma unroll
      for (int r = 0; r < 8; ++r)
        s->costT[gcol][q0 + r + khalf * 8] = dres[r];
    }
  }
  __syncthreads();

  // -------- Jonker-Volgenant Hungarian, block-parallel over 900 columns ----
  const double INF = 1e300;
  for (int j = tid; j < NCOLP; j += NT) {
    s->v[j] = 0.0;
    s->p[j] = 0;
  }
  for (int i = tid; i <= NM; i += NT) s->u[i] = 0.0;
  __syncthreads();

  for (int i = 1; i <= n; ++i) {
    if (tid == 0) {
      s->p[0] = i;
      s->j0 = 0;
    }
    for (int j = tid; j < NCOLP; j += NT) {
      s->minv[j] = INF;
      s->used[j] = 0;
    }
    __syncthreads();
    while (true) {
      if (tid == 0) {
        s->used[s->j0] = 1;
        s->i0 = s->p[s->j0];
      }
      __syncthreads();
      const int i0 = s->i0;
      const int j0c = s->j0;
      const float* crow = s->costT[i0 - 1];
      const double ui0 = s->u[i0];
      double bv = INF;
      int bj = 0x7fffffff;
      for (int j = 1 + tid; j <= NQ; j += NT) {
        if (!s->used[j]) {
          double cur = (double)crow[j - 1] - ui0 - s->v[j];
          double mv = s->minv[j];
          if (cur < mv) {
            mv = cur;
            s->minv[j] = cur;
            s->way[j] = j0c;
          }
          if (mv < bv || (mv == bv && j < bj)) {
            bv = mv;
            bj = j;
          }
        }
      }
      s->red_val[tid] = bv;
      s->red_idx[tid] = bj;
      __syncthreads();
      for (int off = NT / 2; off > 0; off >>= 1) {
        if (tid < off) {
          double ov = s->red_val[tid + off];
          int oj = s->red_idx[tid + off];
          if (ov < s->red_val[tid] ||
              (ov == s->red_val[tid] && oj < s->red_idx[tid])) {
            s->red_val[tid] = ov;
            s->red_idx[tid] = oj;
          }
        }
        __syncthreads();
      }
      const double delta = s->red_val[0];
      const int j1 = s->red_idx[0];
      for (int j = tid; j <= NQ; j += NT) {
        if (s->used[j]) {
          s->u[s->p[j]] += delta;  // p[] injective over used cols: no conflict
          s->v[j] -= delta;
        } else {
          s->minv[j] -= delta;
        }
      }
      __syncthreads();
      if (tid == 0) {
        s->j0 = j1;
        s->done = (s->p[j1] == 0);
      }
      __syncthreads();
      if (s->done) break;
    }
    if (tid == 0) {  // augment along alternating path
      int j0 = s->j0;
      while (j0) {
        int j1 = s->way[j0];
        s->p[j0] = s->p[j1];
        j0 = j1;
      }
    }
    __syncthreads();
  }

  // -------- Loss accumulation over matched pairs ---------------------------
  double ce = 0.0, l1s = 0.0;
  for (int j = 1 + tid; j <= NQ; j += NT) {
    const int r = s->p[j];
    if (r > 0) {
      const int row = r - 1, q = j - 1;
      const int lab = s->labels_c[row];
      ce += -log((double)s->prob[q][lab]);  // CE == -log softmax at label
      double t = 0.0;
      for (int d = 0; d < 6; ++d)
        t += fabs((double)s->pb[q][d] - (double)s->gb[row][d]);
      l1s += t;
    }
  }
  s->red_val[tid] = ce;
  __syncthreads();
  for (int off = NT / 2; off > 0; off >>= 1) {
    if (tid < off) s->red_val[tid] += s->red_val[tid + off];
    __syncthreads();
  }
  const double ce_tot = s->red_val[0];
  __syncthreads();
  s->red_val[tid] = l1s;
  __syncthreads();
  for (int off = NT / 2; off > 0; off >>= 1) {
    if (tid < off) s->red_val[tid] += s->red_val[tid + off];
    __syncthreads();
  }
  if (tid == 0) {
    atomicAdd(&acc[0], ce_tot);
    atomicAdd(&acc[1], s->red_val[0]);
    atomicAdd(&acc[2], (double)n);
  }
}

__global__ void finalize_kernel(const double* __restrict__ acc,
                                float* __restrict__ out) {
  const double N = acc[2];
  const double cl = acc[0] / N;
  const double bl = acc[1] / (N * 6.0);
  out[0] = (float)cl;
  out[1] = (float)bl;
  out[2] = (float)(cl + 5.0 * bl);  // W_CLASS=1, W_BOX=5
}

extern "C" void kernel_launch(void* const* d_in, const int* in_sizes, int n_in,
                              void* d_out, int out_size, void* d_ws,
                              size_t ws_size, hipStream_t stream) {
  const float* pred_boxes = (const float*)d_in[0];
  const float* pred_classes = (const float*)d_in[1];
  const float* gt_boxes = (const float*)d_in[2];
  const int* gt_labels = (const int*)d_in[3];
  const int* gt_valid = (const int*)d_in[4];
  double* acc = (double*)d_ws;  // [ce_sum, l1_sum, pair_count] as doubles
  float* out = (float*)d_out;

  init_acc_kernel<<<1, 1, 0, stream>>>(acc);
  const size_t smem = sizeof(SM);  // ~292 KB, fits CDNA5 320KB-per-WG LDS
  match_loss_kernel<<<NB, NT, smem, stream>>>(pred_boxes, pred_classes,
                                              gt_boxes, gt_labels, gt_valid,
                                              acc);
  finalize_kernel<<<1, 1, 0, stream>>>(acc, out);
}